// LArEnnet_88940182766342
// MI455X (gfx1250) — compile-verified
//
#include <hip/hip_runtime.h>
#include <math.h>

typedef __attribute__((ext_vector_type(2))) float v2f;
typedef __attribute__((ext_vector_type(8))) float v8f;

#define WAVES_PER_BLOCK 8
#define TPB (WAVES_PER_BLOCK * 32)

// smooth unit step: exp(-1/x) for x>0 else 0
__device__ __forceinline__ float sus_f(float x) {
    return x > 0.0f ? expf(-1.0f / x) : 0.0f;
}

// ---------------------------------------------------------------------------
// Kernel 1: per-edge radial embedding  emb[e][0..2]
// soft_one_hot_linspace(d,0,3,3,'smooth_finite',cutoff)*sqrt(3)
// ---------------------------------------------------------------------------
__global__ void lar_emb_kernel(const float* __restrict__ pos,
                               const int* __restrict__ esrc,
                               const int* __restrict__ edst,
                               float* __restrict__ emb, int E) {
    int e = blockIdx.x * blockDim.x + threadIdx.x;
    if (e >= E) return;
    int s = esrc[e], d = edst[e];
    float dx = pos[d * 3 + 0] - pos[s * 3 + 0];
    float dy = pos[d * 3 + 1] - pos[s * 3 + 1];
    float dz = pos[d * 3 + 2] - pos[s * 3 + 2];
    float r = sqrtf(dx * dx + dy * dy + dz * dz);
    const float STEP = 0.75f;                       // max_radius/(nb+1)
    const float PREF = 8.4335731f * 1.7320508f;     // C_SMOOTH * sqrt(num_basis)
#pragma unroll
    for (int i = 0; i < 3; ++i) {
        float v = 0.75f * (float)(i + 1);           // linspace(0,3,5)[1:-1]
        float t = (r - v) / STEP;
        emb[e * 3 + i] = PREF * sus_f(t + 1.0f) * sus_f(1.0f - t);
    }
}

// ---------------------------------------------------------------------------
// Conv layer: per wave, 16-edge tile.
//   h[e,0:16]   = relu(emb[e] @ W1/sqrt(3))                  (VALU, K=3)
//   msg[e,ow]   = sum_{u,k} (h[e,k]*xsrc[e,u]) * Bs[(u,k),ow] (f32 WMMA chain)
// where Bs[(u,k),ow] = w2[k, u*MOUT+ow] * (sqrt2*path_w/16)
// Result accumulated into xout[dst] with global f32 atomics.
// ---------------------------------------------------------------------------
template <int MIN_, int MOUT_>
__global__ void lar_conv_kernel(const float* __restrict__ emb,
                                const float* __restrict__ xin,
                                const int* __restrict__ esrc,
                                const int* __restrict__ edst,
                                const float* __restrict__ w1,
                                const float* __restrict__ w2, int w2ld,
                                float w2scale,
                                float* __restrict__ xout, int E) {
    __shared__ float W1s[3][16];
    __shared__ float W2s[16][72];                       // padded: halves 16 banks apart
    __shared__ float hS[WAVES_PER_BLOCK][16][17];       // padded: conflict-free column read
    __shared__ float xsS[WAVES_PER_BLOCK][16][5];
    __shared__ int dstS[WAVES_PER_BLOCK][16];

    const int tid = threadIdx.x;
    const float inv_sqrt3 = 0.57735027f;
    for (int i = tid; i < 48; i += TPB)
        W1s[i >> 4][i & 15] = w1[i] * inv_sqrt3;
    for (int i = tid; i < 16 * 72; i += TPB) {
        int k = i / 72, j = i % 72;
        W2s[k][j] = (j < MIN_ * MOUT_) ? w2[k * w2ld + j] * w2scale : 0.0f;
    }
    __syncthreads();

    const int lane = tid & 31;
    const int wv = tid >> 5;
    const int hi = lane >> 4;     // half-wave select: K offset 0 or 2
    const int row = lane & 15;    // edge row (A) / output column (B,C)

    // B fragments: constant across tiles -> hoist into registers.
    v2f bfrag[MIN_ * 4];
#pragma unroll
    for (int c = 0; c < MIN_ * 4; ++c) {
        int u = c >> 2;
        int kb = ((c & 3) << 2) + (hi << 1);
        float b0 = (row < MOUT_) ? W2s[kb][u * MOUT_ + row] : 0.0f;
        float b1 = (row < MOUT_) ? W2s[kb + 1][u * MOUT_ + row] : 0.0f;
        bfrag[c].x = b0;
        bfrag[c].y = b1;
    }

    const int ntiles = (E + 15) >> 4;
    const int gw = blockIdx.x * WAVES_PER_BLOCK + wv;
    const int nw = gridDim.x * WAVES_PER_BLOCK;

    for (int tile = gw; tile < ntiles; tile += nw) {
        // ---- stage phase: lanes 0..15 each own one edge ----
        if (lane < 16) {
            int eg = tile * 16 + lane;
            float e0 = 0.0f, e1 = 0.0f, e2 = 0.0f;
            int dd = -1;
            float xs[MIN_];
#pragma unroll
            for (int u = 0; u < MIN_; ++u) xs[u] = 0.0f;
            if (eg < E) {
                int s = esrc[eg];
                dd = edst[eg];
                e0 = emb[eg * 3 + 0];
                e1 = emb[eg * 3 + 1];
                e2 = emb[eg * 3 + 2];
#pragma unroll
                for (int u = 0; u < MIN_; ++u) xs[u] = xin[s * MIN_ + u];
            }
            dstS[wv][lane] = dd;
#pragma unroll
            for (int u = 0; u < MIN_; ++u) xsS[wv][lane][u] = xs[u];
#pragma unroll
            for (int j = 0; j < 16; ++j) {
                float h = e0 * W1s[0][j] + e1 * W1s[1][j] + e2 * W1s[2][j];
                hS[wv][lane][j] = fmaxf(h, 0.0f);
            }
        }
        // wave-local LDS handoff (lanes 16..31 read lanes 0..15's stores)
        asm volatile("s_wait_dscnt 0x0" ::: "memory");

        // Hoist destination indices now: these ds loads are independent of the
        // WMMA chain below, so they issue early and overlap the matrix ops,
        // leaving the scatter tail free of LDS waits.
        int dstv[8];
#pragma unroll
        for (int j = 0; j < 8; ++j) dstv[j] = dstS[wv][j + (hi << 3)];

        // ---- WMMA phase: full EXEC, chained f32 16x16x4 ----
        v8f acc = {};
        float xsv = 0.0f;
#pragma unroll
        for (int c = 0; c < MIN_ * 4; ++c) {
            int u = c >> 2;
            int kb = ((c & 3) << 2) + (hi << 1);
            if ((c & 3) == 0) xsv = xsS[wv][row][u];
            v2f a;
            a.x = hS[wv][row][kb] * xsv;
            a.y = hS[wv][row][kb + 1] * xsv;
            acc = __builtin_amdgcn_wmma_f32_16x16x4_f32(
                false, a, false, bfrag[c], (short)0, acc, false, false);
        }

        // ---- scatter: C layout VGPR j -> rows j / j+8, col = lane&15 ----
        if (row < MOUT_) {
#pragma unroll
            for (int j = 0; j < 8; ++j) {
                if (dstv[j] >= 0) {
                    atomicAdd(&xout[dstv[j] * MOUT_ + row], acc[j]);
                }
            }
        }
    }
}

// ---------------------------------------------------------------------------
// Head: out[n] = x16[n] @ conv_w.T + conv_b   (16 -> 2)
// ---------------------------------------------------------------------------
__global__ void lar_head_kernel(const float* __restrict__ x16,
                                const float* __restrict__ cw,
                                const float* __restrict__ cb,
                                float* __restrict__ out, int N) {
    int n = blockIdx.x * blockDim.x + threadIdx.x;
    if (n >= N) return;
    float s0 = cb[0], s1 = cb[1];
#pragma unroll
    for (int j = 0; j < 16; ++j) {
        float v = x16[n * 16 + j];
        s0 += v * cw[j];
        s1 += v * cw[16 + j];
    }
    out[n * 2 + 0] = s0;
    out[n * 2 + 1] = s1;
}

// ---------------------------------------------------------------------------
extern "C" void kernel_launch(void* const* d_in, const int* in_sizes, int n_in,
                              void* d_out, int out_size, void* d_ws, size_t ws_size,
                              hipStream_t stream) {
    (void)n_in; (void)out_size; (void)ws_size;
    const float* pos     = (const float*)d_in[0];
    const float* feat    = (const float*)d_in[1];
    const int*   esrc    = (const int*)d_in[2];
    const int*   edst    = (const int*)d_in[3];
    const float* fc1_w1  = (const float*)d_in[4];
    const float* fc1_w2  = (const float*)d_in[5];
    const float* core_w1 = (const float*)d_in[6];
    const float* core_w2 = (const float*)d_in[7];
    const float* fc3_w1  = (const float*)d_in[8];
    const float* fc3_w2  = (const float*)d_in[9];
    const float* conv_w  = (const float*)d_in[10];
    const float* conv_b  = (const float*)d_in[11];
    float* out = (float*)d_out;

    const int N = in_sizes[0] / 3;
    const int E = in_sizes[2];

    float* emb = (float*)d_ws;
    float* xA  = emb + (size_t)E * 3;
    float* xB  = xA + (size_t)N * 4;
    float* x16 = xB + (size_t)N * 4;

    lar_emb_kernel<<<(E + 255) / 256, 256, 0, stream>>>(pos, esrc, edst, emb, E);

    // w2scale = sqrt(2) * path_w / (4 [w2 norm] * 4 [1/sqrt(z)])
    const float sc1 = 1.41421356f * 0.57735027f / 16.0f;  // conv1: pw = 1/sqrt(3)
    const float scc = 1.41421356f * 0.5f / 16.0f;         // core & conv3: pw = 0.5

    const int convBlocks = 1280;

    hipMemsetAsync(xA, 0, (size_t)N * 4 * sizeof(float), stream);
    lar_conv_kernel<3, 4><<<convBlocks, TPB, 0, stream>>>(
        emb, feat, esrc, edst, fc1_w1, fc1_w2, 12, sc1, xA, E);

    float* xi = xA;
    float* xo = xB;
    for (int i = 0; i < 3; ++i) {
        hipMemsetAsync(xo, 0, (size_t)N * 4 * sizeof(float), stream);
        lar_conv_kernel<4, 4><<<convBlocks, TPB, 0, stream>>>(
            emb, xi, esrc, edst, core_w1 + (size_t)i * 48,
            core_w2 + (size_t)i * 768, 48, scc, xo, E);
        float* t = xi; xi = xo; xo = t;
    }

    hipMemsetAsync(x16, 0, (size_t)N * 16 * sizeof(float), stream);
    lar_conv_kernel<4, 16><<<convBlocks, TPB, 0, stream>>>(
        emb, xi, esrc, edst, fc3_w1, fc3_w2, 64, scc, x16, E);

    lar_head_kernel<<<(N + 255) / 256, 256, 0, stream>>>(x16, conv_w, conv_b, out, N);
}